// RhythmMamba_8985071583557
// MI455X (gfx1250) — compile-verified
//
#include <hip/hip_runtime.h>
#include <math.h>

// ---------------------------------------------------------------------------
// RhythmMamba forward for MI455X (gfx1250, wave32).
// All matrix work uses V_WMMA_F32_16X16X4_F32 (full f32 precision; model is
// f32 and bandwidth/latency bound at these sizes, so low-precision WMMA buys
// nothing). One wave per 16x16 C tile => EXEC all-ones at every WMMA.
// Inner loops are branchless: OOB rows/cols of a WMMA tile only affect C
// entries that are never stored, so A/B loads run unguarded on clamped
// pointers; conv zero-padding uses clamped loads + selects; im2col K-decode
// uses incremental carry counters (no integer division in the hot loop).
// ---------------------------------------------------------------------------

#define EMBED   96
#define D_INNER 192
#define D_STATE 48
#define DT_RANK 6
#define TLEN    80
#define NDEP    160
#define NB      2
#define DBL_LD  112   // padded row stride for the (.,102) x_proj output

typedef float v2f __attribute__((ext_vector_type(2)));
typedef float v8f __attribute__((ext_vector_type(8)));

__device__ __forceinline__ v8f wmma_f32_16x16x4(v2f a, v2f b, v8f c) {
  // (neg_a, A, neg_b, B, c_mod, C, reuse_a, reuse_b)
  return __builtin_amdgcn_wmma_f32_16x16x4_f32(false, a, false, b, (short)0, c,
                                               false, false);
}

__device__ __forceinline__ int imin(int a, int b) { return a < b ? a : b; }
__device__ __forceinline__ int imax(int a, int b) { return a > b ? a : b; }
__device__ __forceinline__ float sigm(float v) { return 1.f / (1.f + __expf(-v)); }

// ---------------------------------------------------------------------------
// Generic WMMA GEMM:  C[M,N] = A[M,K] * W[N,K]^T  (+ epilogue)
// K is a compile-time constant (multiple of 4) -> fully branchless inner loop
// of unconditional b64 loads + WMMA.  flags bit0 = relu, bit1 = add-to-C.
// epilogue: v = (acc + bias[n]) * scale[n] + shift[n]
// ---------------------------------------------------------------------------
template <int K>
__global__ void gemm_wmma_kernel(const float* __restrict__ A,
                                 const float* __restrict__ W,
                                 float* __restrict__ C, int M, int N, int lda,
                                 int ldb, int ldc,
                                 const float* __restrict__ bias,
                                 const float* __restrict__ scale,
                                 const float* __restrict__ shift, int flags) {
  int m0 = blockIdx.x << 4, n0 = blockIdx.y << 4;
  int lane = threadIdx.x;
  int hf = lane >> 4, q = lane & 15;
  int mA = imin(m0 + q, M - 1);  // clamp: OOB rows only feed unsaved C rows
  int nB = imin(n0 + q, N - 1);  // clamp: OOB cols only feed unsaved C cols
  const float* Arow = A + (size_t)mA * lda + 2 * hf;
  const float* Wrow = W + (size_t)nB * ldb + 2 * hf;
  v8f acc = {};
#pragma unroll 4
  for (int k0 = 0; k0 < K; k0 += 4) {
    __builtin_prefetch(Arow + k0 + 64, 0, 0);  // global_prefetch_b8
    v2f a = *(const v2f*)(Arow + k0);          // global_load_b64
    v2f b = *(const v2f*)(Wrow + k0);          // global_load_b64
    acc = wmma_f32_16x16x4(a, b, acc);
  }
  int col = n0 + q;
  if (col < N) {
    float bs = bias ? bias[col] : 0.f;
    float sc = scale ? scale[col] : 1.f;
    float sh = shift ? shift[col] : 0.f;
    int rbase = m0 + (hf ? 8 : 0);
#pragma unroll
    for (int r = 0; r < 8; ++r) {
      int row = rbase + r;
      if (row < M) {
        float v = (acc[r] + bs) * sc + sh;
        if (flags & 1) v = fmaxf(v, 0.f);
        float* p = C + (size_t)row * ldc + col;
        if (flags & 2) v += *p;
        *p = v;
      }
    }
  }
}

// ---------------------------------------------------------------------------
// 7x7 conv (pad 3) as implicit GEMM.  M = Hout*Wout pixels, N = Cout, K=CIN*49
// DIFF: A operand is the fusion-stem frame difference computed on the fly
// from x (never materializing the 250 MB diff tensor).
// K-decode (cin,kh,kw) kept in incremental carry counters: each lane's k
// advances by exactly 4 per WMMA step, so each carry fires at most once.
// Epilogue: relu((acc + bias) * g + be)
// ---------------------------------------------------------------------------
template <int CIN, int STRIDE, bool DIFF>
__global__ void conv7_wmma_kernel(const float* __restrict__ in,
                                  const float* __restrict__ xsrc,
                                  const float* __restrict__ Wt,
                                  float* __restrict__ out,
                                  const float* __restrict__ bias,
                                  const float* __restrict__ gam,
                                  const float* __restrict__ bet, int Hin,
                                  int Win, int Cout, int Wout, int HW) {
  const int K = CIN * 49;
  int img = blockIdx.z;
  int m0 = blockIdx.x << 4, n0 = blockIdx.y << 4;
  int lane = threadIdx.x;
  int hf = lane >> 4, q = lane & 15;
  int m = m0 + q;  // HW is always a multiple of 16 here
  int oh = m / Wout, ow = m - oh * Wout;
  int ohs = oh * STRIDE - 3, ows = ow * STRIDE - 3;
  int n = imin(n0 + q, Cout - 1);  // clamped; OOB cols never stored
  const float* Wrow = Wt + (size_t)n * K;
  int nimg = img / NDEP, dimg = img - nimg * NDEP;
  size_t inBase = (size_t)img * CIN * Hin * Win;
  size_t dfBase = (size_t)nimg * NDEP * 3 * Hin * Win;
  int chw = 3 * Hin * Win;  // frame stride for diff mode
  // per-lane incremental K-decode counters (t=0,1 handle k and k+1)
  int kw[2], kh[2], cin[2], kb[2], cc[2];
#pragma unroll
  for (int t = 0; t < 2; ++t) {
    kw[t] = 2 * hf + t;  // initial k <= 3  =>  cin=kh=0, kw=k
    kh[t] = 0; cin[t] = 0; kb[t] = 0; cc[t] = 0;
  }
  v8f acc = {};
  for (int k0 = 0; k0 < K; k0 += 4) {
    float av[2];
#pragma unroll
    for (int t = 0; t < 2; ++t) {
      int ih = ohs + kh[t], iw = ows + kw[t];
      bool inb = (ih >= 0) & (ih < Hin) & (iw >= 0) & (iw < Win) & (cin[t] < CIN);
      int ihc = imin(imax(ih, 0), Hin - 1);
      int iwc = imin(imax(iw, 0), Win - 1);
      float v;
      if (!DIFF) {
        int cinc = imin(cin[t], CIN - 1);
        v = in[inBase + ((size_t)cinc * Hin + ihc) * Win + iwc];
      } else {
        // diff block kb (0..3), channel cc: x[g(kb-1)] - x[g(kb-2)]
        int ga = imin(imax(dimg + kb[t] - 2, 0), NDEP - 1);
        int gb = imin(imax(dimg + kb[t] - 1, 0), NDEP - 1);
        size_t sp = ((size_t)cc[t] * Hin + ihc) * Win + iwc;
        v = xsrc[dfBase + (size_t)gb * chw + sp] -
            xsrc[dfBase + (size_t)ga * chw + sp];
      }
      av[t] = inb ? v : 0.f;  // v_cndmask, no exec manipulation
    }
    v2f a; a.x = av[0]; a.y = av[1];
    v2f b;
    if (K % 4 == 0) {
      b = *(const v2f*)(Wrow + k0 + 2 * hf);  // unconditional b64
    } else {  // s11 tail (K=147)
      int kb2 = k0 + 2 * hf;
      float b0 = Wrow[imin(kb2, K - 1)];
      float b1 = Wrow[imin(kb2 + 1, K - 1)];
      b.x = (kb2 < K) ? b0 : 0.f;
      b.y = (kb2 + 1 < K) ? b1 : 0.f;
    }
    acc = wmma_f32_16x16x4(a, b, acc);
    // advance counters by +4 (at most one carry per level)
#pragma unroll
    for (int t = 0; t < 2; ++t) {
      kw[t] += 4;
      bool c1 = kw[t] >= 7;
      kw[t] = c1 ? kw[t] - 7 : kw[t];
      kh[t] = c1 ? kh[t] + 1 : kh[t];
      bool c2 = kh[t] >= 7;
      kh[t] = c2 ? kh[t] - 7 : kh[t];
      cin[t] = c2 ? cin[t] + 1 : cin[t];
      if (DIFF) {
        cc[t] = c2 ? cc[t] + 1 : cc[t];
        bool c3 = cc[t] >= 3;
        cc[t] = c3 ? 0 : cc[t];
        kb[t] = c3 ? kb[t] + 1 : kb[t];
      }
    }
  }
  int col = n0 + q;
  if (col < Cout) {
    float bs = bias[col], gm = gam[col], bt = bet[col];
    int rbase = m0 + (hf ? 8 : 0);
    size_t obase = ((size_t)img * Cout + col) * (size_t)HW;
#pragma unroll
    for (int r = 0; r < 8; ++r)
      out[obase + rbase + r] = fmaxf((acc[r] + bs) * gm + bt, 0.f);
  }
}

// ---------------------------------------------------------------------------
// stem3: 3D conv (96,24,2,5,5), stride (2,1,1), pad (0,2,2), + bias + BN.
// Per (b,dd): GEMM  M=256 pixels, N=96, K=24*2*25=1200 (multiple of 4).
// Incremental (ci,kd,kh,kw) carry counters, same scheme as conv7.
// ---------------------------------------------------------------------------
__global__ void stem3_wmma_kernel(const float* __restrict__ fin,
                                  const float* __restrict__ Wt,
                                  const float* __restrict__ bias,
                                  const float* __restrict__ gam,
                                  const float* __restrict__ bet,
                                  float* __restrict__ out) {
  int bz = blockIdx.z;  // b*80 + dd
  int b = bz / TLEN, dd = bz - b * TLEN;
  int m0 = blockIdx.x << 4, n0 = blockIdx.y << 4;
  int lane = threadIdx.x;
  int hf = lane >> 4, q = lane & 15;
  const int K = 24 * 2 * 25;  // 1200
  int m = m0 + q;
  int oh = m >> 4, ow = m & 15;
  int n = n0 + q;  // < 96 always
  const float* Wrow = Wt + (size_t)n * K + 2 * hf;
  size_t finBase = (size_t)b * NDEP * 24 * 256;
  int dd2 = 2 * dd;
  int kw[2], kh[2], kd[2], ci[2];
#pragma unroll
  for (int t = 0; t < 2; ++t) {
    kw[t] = 2 * hf + t;  // initial k <= 3 => kw=k (<5), rest 0
    kh[t] = 0; kd[t] = 0; ci[t] = 0;
  }
  v8f acc = {};
  for (int k0 = 0; k0 < K; k0 += 4) {
    float av[2];
#pragma unroll
    for (int t = 0; t < 2; ++t) {
      int din = dd2 + kd[t];  // 0..159, no pad in D
      int ih = oh + kh[t] - 2, iw = ow + kw[t] - 2;
      bool inb = (ih >= 0) & (ih < 16) & (iw >= 0) & (iw < 16);
      int ihc = imin(imax(ih, 0), 15);
      int iwc = imin(imax(iw, 0), 15);
      float v = fin[finBase + ((size_t)din * 24 + ci[t]) * 256 + ihc * 16 + iwc];
      av[t] = inb ? v : 0.f;
    }
    v2f a; a.x = av[0]; a.y = av[1];
    v2f bf = *(const v2f*)(Wrow + k0);  // global_load_b64, unconditional
    acc = wmma_f32_16x16x4(a, bf, acc);
#pragma unroll
    for (int t = 0; t < 2; ++t) {
      kw[t] += 4;
      bool c1 = kw[t] >= 5;
      kw[t] = c1 ? kw[t] - 5 : kw[t];
      kh[t] = c1 ? kh[t] + 1 : kh[t];
      bool c2 = kh[t] >= 5;
      kh[t] = c2 ? kh[t] - 5 : kh[t];
      kd[t] = c2 ? kd[t] + 1 : kd[t];
      bool c3 = kd[t] >= 2;
      kd[t] = c3 ? kd[t] - 2 : kd[t];
      ci[t] = c3 ? ci[t] + 1 : ci[t];
    }
  }
  int col = n0 + q;
  float bs = bias[col], gm = gam[col], bt = bet[col];
  int rbase = m0 + (hf ? 8 : 0);
  size_t obase = (((size_t)b * EMBED + col) * TLEN + dd) * 256;
#pragma unroll
  for (int r = 0; r < 8; ++r) out[obase + rbase + r] = (acc[r] + bs) * gm + bt;
}

// ----------------------------- elementwise helpers -------------------------
__global__ void maxpool2_kernel(const float* __restrict__ in,
                                float* __restrict__ out, int NC, int Ho,
                                int Wo) {
  int idx = blockIdx.x * blockDim.x + threadIdx.x;
  int total = NC * Ho * Wo;
  if (idx >= total) return;
  int ow = idx % Wo;
  int oh = (idx / Wo) % Ho;
  int c = idx / (Wo * Ho);
  const float* p = in + (((size_t)c * Ho * 2) + oh * 2) * (Wo * 2) + ow * 2;
  out[idx] = fmaxf(fmaxf(p[0], p[1]), fmaxf(p[2 * Wo], p[2 * Wo + 1]));
}

__global__ void avg_kernel(const float* __restrict__ a,
                           const float* __restrict__ b, float* __restrict__ o,
                           int n) {
  int i = blockIdx.x * blockDim.x + threadIdx.x;
  if (i < n) o[i] = 0.5f * (a[i] + b[i]);
}

// sigmoid-mask pooled mean: h[b,t,c] = 0.5 * sum(v*sig(v)) / sum(sig(v))
__global__ void stem3_pool_kernel(const float* __restrict__ f3,
                                  float* __restrict__ h) {
  __shared__ float ss[256], sv[256];
  int blk = blockIdx.x;  // (b*96 + c)*80 + t
  int tid = threadIdx.x;
  float v = f3[(size_t)blk * 256 + tid];
  float sg = sigm(v);
  ss[tid] = sg;
  sv[tid] = v * sg;
  __syncthreads();
  for (int off = 128; off; off >>= 1) {
    if (tid < off) { ss[tid] += ss[tid + off]; sv[tid] += sv[tid + off]; }
    __syncthreads();
  }
  if (tid == 0) {
    int t = blk % 80;
    int c = (blk / 80) % 96;
    int b = blk / (80 * 96);
    h[((size_t)b * TLEN + t) * EMBED + c] = 0.5f * sv[0] / ss[0];
  }
}

// ------------------------------- LayerNorms --------------------------------
// gather the 4 shifted copies of x and LayerNorm (aln) -> xin (8,80,96)
__global__ void ln_gather_kernel(const float* __restrict__ hsrc,
                                 const float* __restrict__ g,
                                 const float* __restrict__ b,
                                 float* __restrict__ out) {
  __shared__ float sm[128], sq[128];
  int blk = blockIdx.x;  // 0..639
  int bidx = blk / TLEN, t = blk % TLEN;
  int i = bidx >> 1, borig = bidx & 1;
  int src_t = (t < TLEN - i * 20) ? (i * 20 + t) : t;
  int c = threadIdx.x;
  float v = (c < EMBED) ? hsrc[((size_t)borig * TLEN + src_t) * EMBED + c] : 0.f;
  sm[c] = v;
  sq[c] = v * v;
  __syncthreads();
  for (int off = 64; off; off >>= 1) {
    if (c < off) { sm[c] += sm[c + off]; sq[c] += sq[c + off]; }
    __syncthreads();
  }
  float mean = sm[0] * (1.f / EMBED);
  float var = sq[0] * (1.f / EMBED) - mean * mean;
  float inv = rsqrtf(var + 1e-5f);
  if (c < EMBED) out[(size_t)blk * EMBED + c] = (v - mean) * inv * g[c] + b[c];
}

// segment recombine + LN(ln1) + residual add into x (in place)
__global__ void combine_ln_res_kernel(const float* __restrict__ ym,
                                      const float* __restrict__ g,
                                      const float* __restrict__ b,
                                      float* __restrict__ x) {
  __shared__ float sm[128], sq[128];
  int blk = blockIdx.x;  // 0..159
  int bb = blk / TLEN, t = blk % TLEN;
  int i = t / 20;
  int c = threadIdx.x;
  float v = 0.f;
  if (c < EMBED) {
    v = ym[((size_t)bb * TLEN + t) * EMBED + c];
    for (int j = 1; j <= i; ++j)
      v += ym[(((size_t)(2 * j + bb)) * TLEN + (t - 20 * j)) * EMBED + c];
    v /= (float)(i + 1);
  }
  sm[c] = v;
  sq[c] = v * v;
  __syncthreads();
  for (int off = 64; off; off >>= 1) {
    if (c < off) { sm[c] += sm[c + off]; sq[c] += sq[c + off]; }
    __syncthreads();
  }
  float mean = sm[0] * (1.f / EMBED);
  float var = sq[0] * (1.f / EMBED) - mean * mean;
  float inv = rsqrtf(var + 1e-5f);
  if (c < EMBED)
    x[(size_t)blk * EMBED + c] += (v - mean) * inv * g[c] + b[c];
}

__global__ void ln_kernel(const float* __restrict__ xin,
                          const float* __restrict__ g,
                          const float* __restrict__ b, float* __restrict__ out) {
  __shared__ float sm[128], sq[128];
  int row = blockIdx.x;
  int c = threadIdx.x;
  float v = (c < EMBED) ? xin[(size_t)row * EMBED + c] : 0.f;
  sm[c] = v;
  sq[c] = v * v;
  __syncthreads();
  for (int off = 64; off; off >>= 1) {
    if (c < off) { sm[c] += sm[c + off]; sq[c] += sq[c + off]; }
    __syncthreads();
  }
  float mean = sm[0] * (1.f / EMBED);
  float var = sq[0] * (1.f / EMBED) - mean * mean;
  float inv = rsqrtf(var + 1e-5f);
  if (c < EMBED) out[(size_t)row * EMBED + c] = (v - mean) * inv * g[c] + b[c];
}

// ------------------------------- Mamba core --------------------------------
// causal depthwise conv1d (k=4, left pad 3) + SiLU; input = xz[:, :192]
__global__ void conv1d_silu_kernel(const float* __restrict__ xz,
                                   const float* __restrict__ cw,
                                   const float* __restrict__ cb,
                                   float* __restrict__ xc) {
  int idx = blockIdx.x * blockDim.x + threadIdx.x;
  if (idx >= 8 * TLEN * D_INNER) return;
  int ch = idx % D_INNER;
  int l = (idx / D_INNER) % TLEN;
  int b = idx / (D_INNER * TLEN);
  float s = cb[ch];
#pragma unroll
  for (int k = 0; k < 4; ++k) {
    int ls = l - 3 + k;
    float v = xz[((size_t)b * TLEN + imax(ls, 0)) * 384 + ch];
    s += (ls >= 0) ? cw[ch * 4 + k] * v : 0.f;
  }
  xc[idx] = s * sigm(s);
}

__global__ void dtproj_kernel(const float* __restrict__ dbl,
                              const float* __restrict__ dtw,
                              const float* __restrict__ dtb,
                              float* __restrict__ dt) {
  int idx = blockIdx.x * blockDim.x + threadIdx.x;
  if (idx >= 640 * D_INNER) return;
  int d = idx % D_INNER;
  int row = idx / D_INNER;
  float s = dtb[d];
#pragma unroll
  for (int r = 0; r < DT_RANK; ++r)
    s += dbl[(size_t)row * DBL_LD + r] * dtw[d * DT_RANK + r];
  dt[idx] = (s > 20.f) ? s : log1pf(__expf(s));  // softplus
}

// selective scan: one block per (b,d), 48 independent state fibers + LDS reduce
__global__ void scan_kernel(const float* __restrict__ dbl,
                            const float* __restrict__ dt,
                            const float* __restrict__ xc,
                            const float* __restrict__ A_log,
                            float* __restrict__ ys) {
  __shared__ float red[64];
  int b = blockIdx.x / D_INNER;
  int d = blockIdx.x % D_INNER;
  int s = threadIdx.x;
  bool act = s < D_STATE;
  float A = act ? -__expf(A_log[d * D_STATE + s]) : 0.f;
  float hst = 0.f;
  for (int l = 0; l < TLEN; ++l) {
    size_t row = (size_t)b * TLEN + l;
    float dtv = dt[row * D_INNER + d];
    float xcv = xc[row * D_INNER + d];
    float Bv = act ? dbl[row * DBL_LD + DT_RANK + s] : 0.f;
    float Cv = act ? dbl[row * DBL_LD + DT_RANK + D_STATE + s] : 0.f;
    hst = __expf(dtv * A) * hst + dtv * Bv * xcv;
    red[threadIdx.x] = act ? hst * Cv : 0.f;
    __syncthreads();
    for (int off = 32; off; off >>= 1) {
      if (threadIdx.x < off) red[threadIdx.x] += red[threadIdx.x + off];
      __syncthreads();
    }
    if (threadIdx.x == 0) ys[row * D_INNER + d] = red[0];
    __syncthreads();
  }
}

// y = (ys + xc*D) * silu(z)
__global__ void gate_kernel(const float* __restrict__ ys,
                            const float* __restrict__ xc,
                            const float* __restrict__ xz,
                            const float* __restrict__ Dp,
                            float* __restrict__ y) {
  int idx = blockIdx.x * blockDim.x + threadIdx.x;
  if (idx >= 640 * D_INNER) return;
  int d = idx % D_INNER;
  int row = idx / D_INNER;
  float zv = xz[(size_t)row * 384 + D_INNER + d];
  y[idx] = (ys[idx] + xc[idx] * Dp[d]) * (zv * sigm(zv));
}

// --------------------------------- FFN DFT ---------------------------------
// forward 80-pt DFT (ortho) + complex diag channel mix + relu on re/im
__global__ void dft_fwd_kernel(const float* __restrict__ h1,
                               const float* __restrict__ rmat,
                               const float* __restrict__ imat,
                               const float* __restrict__ rb,
                               const float* __restrict__ ib,
                               float* __restrict__ xr, float* __restrict__ xi) {
  int idx = blockIdx.x * blockDim.x + threadIdx.x;
  if (idx >= NB * TLEN * D_INNER) return;
  int ch = idx % D_INNER;
  int k = (idx / D_INNER) % TLEN;
  int b = idx / (D_INNER * TLEN);
  const float step = 6.283185307179586f / TLEN;
  float sr = 0.f, si = 0.f;
  for (int n = 0; n < TLEN; ++n) {
    float v = h1[((size_t)b * TLEN + n) * D_INNER + ch];
    int p = (k * n) % TLEN;
    float sn, cs;
    __sincosf(-step * (float)p, &sn, &cs);
    sr += v * cs;
    si += v * sn;
  }
  const float sc = 0.11180339887498949f;  // 1/sqrt(80)
  sr *= sc;
  si *= sc;
  float rd = rmat[(size_t)ch * D_INNER + ch];
  float id = imat[(size_t)ch * D_INNER + ch];
  xr[idx] = fmaxf(sr * rd - si * id + rb[ch], 0.f);
  xi[idx] = fmaxf(si * rd + sr * id + ib[ch], 0.f);
}

// inverse 80-pt DFT (ortho), real part
__global__ void dft_inv_kernel(const float* __restrict__ xr,
                               const float* __restrict__ xi,
                               float* __restrict__ h2) {
  int idx = blockIdx.x * blockDim.x + threadIdx.x;
  if (idx >= NB * TLEN * D_INNER) return;
  int ch = idx % D_INNER;
  int n = (idx / D_INNER) % TLEN;
  int b = idx / (D_INNER * TLEN);
  const float step = 6.283185307179586f / TLEN;
  float s = 0.f;
  for (int k = 0; k < TLEN; ++k) {
    size_t src = ((size_t)b * TLEN + k) * D_INNER + ch;
    int p = (k * n) % TLEN;
    float sn, cs;
    __sincosf(step * (float)p, &sn, &cs);
    s += xr[src] * cs - xi[src] * sn;
  }
  h2[idx] = s * 0.11180339887498949f;
}

// --------------------------------- head ------------------------------------
__global__ void head_kernel(const float* __restrict__ h,
                            const float* __restrict__ lw,
                            const float* __restrict__ lb,
                            float* __restrict__ out) {
  int idx = blockIdx.x * blockDim.x + threadIdx.x;
  if (idx >= NB * 160) return;
  int b = idx / 160, t2 = idx % 160;
  float s = lb[0];
  for (int c = 0; c < EMBED; ++c)
    s += h[((size_t)b * TLEN + (t2 >> 1)) * EMBED + c] * lw[c];
  out[idx] = s;
}

// ===========================================================================
extern "C" void kernel_launch(void* const* d_in, const int* in_sizes, int n_in,
                              void* d_out, int out_size, void* d_ws,
                              size_t ws_size, hipStream_t stream) {
  (void)in_sizes; (void)n_in; (void)out_size; (void)ws_size;
  auto F = [&](int i) { return (const float*)d_in[i]; };
  const float* x = F(0);
  const float *s11_w = F(1), *s11_b = F(2), *s11_g = F(3), *s11_be = F(4);
  const float *s12_w = F(5), *s12_b = F(6), *s12_g = F(7), *s12_be = F(8);
  const float *s21_w = F(9), *s21_b = F(10), *s21_g = F(11), *s21_be = F(12);
  const float *s22_w = F(13), *s22_b = F(14), *s22_g = F(15), *s22_be = F(16);
  const float *st3_w = F(17), *st3_b = F(18), *st3_g = F(19), *st3_be = F(20);
  const float *last_w = F(221), *last_b = F(222);

  float* wsp = (float*)d_ws;
  size_t off = 0;
  auto alloc = [&](size_t n) { float* p = wsp + off; off += n; return p; };
  float* sbuf  = alloc(320ull * 12 * 64 * 64);  // s11 pre-pool; reused as c21
  float* dbuf  = alloc(320ull * 12 * 64 * 64);  // s12 pre-pool; reused as c22
  float* spool = alloc(320ull * 12 * 32 * 32);
  float* dpool = alloc(320ull * 12 * 32 * 32);
  float* p1in  = alloc(320ull * 12 * 32 * 32);
  float* p1p   = alloc(320ull * 24 * 16 * 16);
  float* p2p   = alloc(320ull * 24 * 16 * 16);
  float* fstem = alloc(320ull * 24 * 16 * 16);
  float* f3    = alloc(2ull * 96 * 80 * 256);
  float* hbuf  = alloc(2ull * 80 * 96);
  float* xinln = alloc(8ull * 80 * 96);
  float* xz    = alloc(8ull * 80 * 384);
  float* xc    = alloc(8ull * 80 * 192);
  float* dblb  = alloc(8ull * 80 * DBL_LD);
  float* dtb   = alloc(8ull * 80 * 192);
  float* ysb   = alloc(8ull * 80 * 192);
  float* yg    = alloc(8ull * 80 * 192);
  float* ym    = alloc(8ull * 80 * 96);
  float* xln2  = alloc(2ull * 80 * 96);
  float* h1    = alloc(2ull * 80 * 192);
  float* xrb   = alloc(2ull * 80 * 192);
  float* xib   = alloc(2ull * 80 * 192);
  float* h2    = alloc(2ull * 80 * 192);
  float* c21 = sbuf;  // overlay: sbuf/dbuf dead after pooling
  float* c22 = dbuf;

  dim3 w32(32);
  // ---- fusion stem ----
  conv7_wmma_kernel<3, 2, false><<<dim3(256, 1, 320), w32, 0, stream>>>(
      x, nullptr, s11_w, sbuf, s11_b, s11_g, s11_be, 128, 128, 12, 64, 4096);
  conv7_wmma_kernel<12, 2, true><<<dim3(256, 1, 320), w32, 0, stream>>>(
      nullptr, x, s12_w, dbuf, s12_b, s12_g, s12_be, 128, 128, 12, 64, 4096);
  int nc1 = 320 * 12, np1 = nc1 * 32 * 32;
  maxpool2_kernel<<<(np1 + 255) / 256, 256, 0, stream>>>(sbuf, spool, nc1, 32, 32);
  maxpool2_kernel<<<(np1 + 255) / 256, 256, 0, stream>>>(dbuf, dpool, nc1, 32, 32);
  avg_kernel<<<(np1 + 255) / 256, 256, 0, stream>>>(spool, dpool, p1in, np1);
  conv7_wmma_kernel<12, 1, false><<<dim3(64, 2, 320), w32, 0, stream>>>(
      p1in, nullptr, s21_w, c21, s21_b, s21_g, s21_be, 32, 32, 24, 32, 1024);
  conv7_wmma_kernel<12, 1, false><<<dim3(64, 2, 320), w32, 0, stream>>>(
      dpool, nullptr, s22_w, c22, s22_b, s22_g, s22_be, 32, 32, 24, 32, 1024);
  int nc2 = 320 * 24, np2 = nc2 * 16 * 16;
  maxpool2_kernel<<<(np2 + 255) / 256, 256, 0, stream>>>(c21, p1p, nc2, 16, 16);
  maxpool2_kernel<<<(np2 + 255) / 256, 256, 0, stream>>>(c22, p2p, nc2, 16, 16);
  avg_kernel<<<(np2 + 255) / 256, 256, 0, stream>>>(p1p, p2p, fstem, np2);
  // ---- stem3 + masked pool -> h (2,80,96) ----
  stem3_wmma_kernel<<<dim3(16, 6, 160), w32, 0, stream>>>(fstem, st3_w, st3_b,
                                                          st3_g, st3_be, f3);
  stem3_pool_kernel<<<2 * 96 * 80, 256, 0, stream>>>(f3, hbuf);

  // ---- 8 RhythmMamba blocks ----
  for (int bi = 0; bi < 8; ++bi) {
    int base = 21 + bi * 25;
    const float *aln_g = F(base + 0), *aln_b = F(base + 1);
    const float *ln1_g = F(base + 2), *ln1_b = F(base + 3);
    const float *ln2_g = F(base + 4), *ln2_b = F(base + 5);
    const float *in_proj = F(base + 6), *conv_w = F(base + 7),
                *conv_b = F(base + 8), *x_proj = F(base + 9);
    const float *dt_w = F(base + 10), *dt_b = F(base + 11),
                *A_log = F(base + 12), *Dp = F(base + 13),
                *out_proj = F(base + 14);
    const float *fc1 = F(base + 15), *bn1g = F(base + 16), *bn1b = F(base + 17);
    const float *rmat = F(base + 18), *imat = F(base + 19), *rb = F(base + 20),
                *ib = F(base + 21);
    const float *fc2 = F(base + 22), *bn2g = F(base + 23), *bn2b = F(base + 24);

    ln_gather_kernel<<<640, 128, 0, stream>>>(hbuf, aln_g, aln_b, xinln);
    gemm_wmma_kernel<96><<<dim3(40, 24), w32, 0, stream>>>(
        xinln, in_proj, xz, 640, 384, 96, 96, 384, nullptr, nullptr, nullptr, 0);
    conv1d_silu_kernel<<<(640 * 192 + 255) / 256, 256, 0, stream>>>(xz, conv_w,
                                                                    conv_b, xc);
    gemm_wmma_kernel<192><<<dim3(40, 7), w32, 0, stream>>>(
        xc, x_proj, dblb, 640, 102, 192, 192, DBL_LD, nullptr, nullptr,
        nullptr, 0);
    dtproj_kernel<<<(640 * 192 + 255) / 256, 256, 0, stream>>>(dblb, dt_w, dt_b,
                                                               dtb);
    scan_kernel<<<8 * D_INNER, 64, 0, stream>>>(dblb, dtb, xc, A_log, ysb);
    gate_kernel<<<(640 * 192 + 255) / 256, 256, 0, stream>>>(ysb, xc, xz, Dp, yg);
    gemm_wmma_kernel<192><<<dim3(40, 6), w32, 0, stream>>>(
        yg, out_proj, ym, 640, 96, 192, 192, 96, nullptr, nullptr, nullptr, 0);
    combine_ln_res_kernel<<<160, 128, 0, stream>>>(ym, ln1_g, ln1_b, hbuf);
    ln_kernel<<<160, 128, 0, stream>>>(hbuf, ln2_g, ln2_b, xln2);
    gemm_wmma_kernel<96><<<dim3(10, 12), w32, 0, stream>>>(
        xln2, fc1, h1, 160, 192, 96, 96, 192, nullptr, bn1g, bn1b, 1);
    dft_fwd_kernel<<<(2 * 80 * 192 + 255) / 256, 256, 0, stream>>>(
        h1, rmat, imat, rb, ib, xrb, xib);
    dft_inv_kernel<<<(2 * 80 * 192 + 255) / 256, 256, 0, stream>>>(xrb, xib, h2);
    gemm_wmma_kernel<192><<<dim3(10, 6), w32, 0, stream>>>(
        h2, fc2, hbuf, 160, 96, 192, 192, 96, nullptr, bn2g, bn2b, 2);
  }
  // ---- head ----
  head_kernel<<<2, 160, 0, stream>>>(hbuf, last_w, last_b, (float*)d_out);
}